// MoleculeGNN_18751827214721
// MI455X (gfx1250) — compile-verified
//
#include <hip/hip_runtime.h>
#include <hip/hip_bf16.h>

// ---------------------------------------------------------------------------
// MoleculeGNN (3-layer GCN) for MI455X / gfx1250, fp32 end-to-end.
// GEMMs use V_WMMA_F32_16X16X4_F32; edge aggregation uses GLOBAL_ATOMIC_ADD_F32.
// ---------------------------------------------------------------------------

#define GNN_N 100000
#define GNN_E 1600000
#define GNN_F 64
#define GNN_H 64
#define GNN_G 512

typedef __attribute__((ext_vector_type(2))) float v2f;
typedef __attribute__((ext_vector_type(8))) float v8f;

// ---------------- degree / dinv ----------------
__global__ void gnn_deg_kernel(const int* __restrict__ dst, float* __restrict__ deg, int E) {
    int e = blockIdx.x * blockDim.x + threadIdx.x;
    if (e < E) atomicAdd(&deg[dst[e]], 1.0f);
}

__global__ void gnn_dinv_kernel(float* __restrict__ deg, int N) {
    int i = blockIdx.x * blockDim.x + threadIdx.x;
    if (i < N) deg[i] = rsqrtf(deg[i] + 1.0f);   // deg buffer becomes dinv in place
}

// ---------------- GEMM: H = X @ W  ([N,64] @ [64,64]) ----------------
// Block = 256 threads = 8 waves. Each wave owns one 16x16 output tile.
// Block covers 32 rows x 64 cols; grid.x = N/32 (N divisible by 32).
__global__ void __launch_bounds__(256)
gnn_gemm64_kernel(const float* __restrict__ X, const float* __restrict__ W,
                  float* __restrict__ Hout) {
    __shared__ float Wl[64 * 64];
    const int tid = threadIdx.x;

    // Stage W (16 KB) into LDS: 256 threads x 4 float4 loads
    #pragma unroll
    for (int i = 0; i < 4; ++i) {
        int idx = (tid + i * 256) * 4;
        *(float4*)&Wl[idx] = *(const float4*)&W[idx];
    }
    __syncthreads();

    const int wave = tid >> 5;          // 0..7
    const int lane = tid & 31;
    const int half = lane >> 4;         // 0: lanes 0-15, 1: lanes 16-31
    const int l    = lane & 15;

    const int m0 = blockIdx.x * 32 + (wave >> 2) * 16;  // row tile base
    const int n0 = (wave & 3) * 16;                     // col tile base

    // A fragment source: row = m0+l, K pair base offset = half*2
    const float* xrow = X + (size_t)(m0 + l) * GNN_F + half * 2;
    const int col = n0 + l;

    v8f c = {};
    #pragma unroll
    for (int k0 = 0; k0 < 64; k0 += 4) {
        // A: 16x4 f32 tile. VGPR0/1 hold K = k0+2*half, k0+2*half+1 for this lane.
        v2f a = *(const v2f*)(xrow + k0);
        // B: 4x16 f32 tile. Rows K = k0+2*half, +1; column = n0+l.
        const int kr = k0 + half * 2;
        v2f b;
        b.x = Wl[(kr + 0) * 64 + col];
        b.y = Wl[(kr + 1) * 64 + col];
        c = __builtin_amdgcn_wmma_f32_16x16x4_f32(
                /*neg_a=*/false, a, /*neg_b=*/false, b,
                /*c_mod=*/(short)0, c, /*reuse_a=*/false, /*reuse_b=*/false);
    }

    // D layout: VGPR r -> M = m0 + r + half*8, N = n0 + l
    const int mbase = m0 + half * 8;
    #pragma unroll
    for (int r = 0; r < 8; ++r) {
        Hout[(size_t)(mbase + r) * GNN_H + col] = c[r];
    }
}

// ---------------- edge scatter: agg[dst] += h[src] * dinv[src]*dinv[dst] ---
// One wave per edge: 32 lanes x float2 = full 64-float row (256B coalesced).
__global__ void __launch_bounds__(256)
gnn_scatter_kernel(const float* __restrict__ Hin, float* __restrict__ Agg,
                   const int* __restrict__ src, const int* __restrict__ dst,
                   const float* __restrict__ dinv, int E) {
    long long t = (long long)blockIdx.x * blockDim.x + threadIdx.x;
    int e = (int)(t >> 5);
    int lane = (int)(t & 31);
    if (e >= E) return;
    const int s = src[e];
    const int d = dst[e];
    const float norm = dinv[s] * dinv[d];
    float2 v = *(const float2*)(Hin + (size_t)s * GNN_H + lane * 2);
    float* ap = Agg + (size_t)d * GNN_H + lane * 2;
    atomicAdd(ap + 0, v.x * norm);
    atomicAdd(ap + 1, v.y * norm);
}

// ---------------- combine: out = agg + h*dinv^2 + bias (optional ReLU) -----
// In-place on Agg. Thread per (node, 4 features).
__global__ void __launch_bounds__(256)
gnn_combine_kernel(float* __restrict__ Agg, const float* __restrict__ Hin,
                   const float* __restrict__ dinv, const float* __restrict__ bias,
                   int N, int doRelu) {
    int t = blockIdx.x * blockDim.x + threadIdx.x;
    int node = t >> 4;
    int q = t & 15;
    if (node >= N) return;
    const float di = dinv[node];
    const float d2 = di * di;
    const size_t off = (size_t)node * GNN_H + q * 4;
    float4 a = *(float4*)&Agg[off];
    float4 h = *(const float4*)&Hin[off];
    float4 b = *(const float4*)&bias[q * 4];
    float4 o;
    o.x = a.x + h.x * d2 + b.x;
    o.y = a.y + h.y * d2 + b.y;
    o.z = a.z + h.z * d2 + b.z;
    o.w = a.w + h.w * d2 + b.w;
    if (doRelu) {
        o.x = fmaxf(o.x, 0.0f); o.y = fmaxf(o.y, 0.0f);
        o.z = fmaxf(o.z, 0.0f); o.w = fmaxf(o.w, 0.0f);
    }
    *(float4*)&Agg[off] = o;
}

// ---------------- global mean pool ----------------
__global__ void __launch_bounds__(256)
gnn_pool_kernel(const float* __restrict__ Hin, const int* __restrict__ batch,
                float* __restrict__ sums, float* __restrict__ cnt, int N) {
    long long t = (long long)blockIdx.x * blockDim.x + threadIdx.x;
    int node = (int)(t >> 5);
    int lane = (int)(t & 31);
    if (node >= N) return;
    const int g = batch[node];
    float2 v = *(const float2*)(Hin + (size_t)node * GNN_H + lane * 2);
    float* sp = sums + (size_t)g * GNN_H + lane * 2;
    atomicAdd(sp + 0, v.x);
    atomicAdd(sp + 1, v.y);
    if (lane == 0) atomicAdd(&cnt[g], 1.0f);
}

__global__ void gnn_final_kernel(const float* __restrict__ sums,
                                 const float* __restrict__ cnt,
                                 float* __restrict__ out, int G) {
    int t = blockIdx.x * blockDim.x + threadIdx.x;
    int g = t >> 6;
    int f = t & 63;
    if (g >= G) return;
    out[(size_t)g * GNN_H + f] = sums[(size_t)g * GNN_H + f] / fmaxf(cnt[g], 1.0f);
}

// ---------------------------------------------------------------------------
extern "C" void kernel_launch(void* const* d_in, const int* in_sizes, int n_in,
                              void* d_out, int out_size, void* d_ws, size_t ws_size,
                              hipStream_t stream) {
    (void)in_sizes; (void)n_in; (void)out_size; (void)ws_size;

    const float* x     = (const float*)d_in[0];
    const int*   ei    = (const int*)d_in[1];
    const int*   batch = (const int*)d_in[2];
    const float* W1 = (const float*)d_in[3];
    const float* b1 = (const float*)d_in[4];
    const float* W2 = (const float*)d_in[5];
    const float* b2 = (const float*)d_in[6];
    const float* W3 = (const float*)d_in[7];
    const float* b3 = (const float*)d_in[8];
    const int* src = ei;            // edge_index[0]
    const int* dst = ei + GNN_E;    // edge_index[1]

    // Workspace layout (floats)
    float* ws   = (float*)d_ws;
    float* dinv = ws;                                   // N (deg -> dinv in place)
    float* bufA = dinv + GNN_N;                         // N*64  (gemm output h)
    float* bufB = bufA + (size_t)GNN_N * GNN_H;         // N*64  (agg / layer output)
    float* sums = bufB + (size_t)GNN_N * GNN_H;         // G*64
    float* cnt  = sums + (size_t)GNN_G * GNN_H;         // G

    const int TB = 256;
    const dim3 blk(TB);
    const size_t featBytes = (size_t)GNN_N * GNN_H * sizeof(float);

    // ---- degrees / normalization ----
    hipMemsetAsync(dinv, 0, GNN_N * sizeof(float), stream);
    gnn_deg_kernel<<<dim3((GNN_E + TB - 1) / TB), blk, 0, stream>>>(dst, dinv, GNN_E);
    gnn_dinv_kernel<<<dim3((GNN_N + TB - 1) / TB), blk, 0, stream>>>(dinv, GNN_N);

    const dim3 gemmGrid(GNN_N / 32);                                  // 3125
    const dim3 scatGrid((unsigned)(((long long)GNN_E * 32 + TB - 1) / TB));
    const dim3 combGrid((GNN_N * 16 + TB - 1) / TB);

    // ---- layer 1: relu(gcn(x, W1, b1)) -> bufB ----
    gnn_gemm64_kernel<<<gemmGrid, blk, 0, stream>>>(x, W1, bufA);
    hipMemsetAsync(bufB, 0, featBytes, stream);
    gnn_scatter_kernel<<<scatGrid, blk, 0, stream>>>(bufA, bufB, src, dst, dinv, GNN_E);
    gnn_combine_kernel<<<combGrid, blk, 0, stream>>>(bufB, bufA, dinv, b1, GNN_N, 1);

    // ---- layer 2: relu(gcn(bufB, W2, b2)) -> bufB ----
    gnn_gemm64_kernel<<<gemmGrid, blk, 0, stream>>>(bufB, W2, bufA);
    hipMemsetAsync(bufB, 0, featBytes, stream);
    gnn_scatter_kernel<<<scatGrid, blk, 0, stream>>>(bufA, bufB, src, dst, dinv, GNN_E);
    gnn_combine_kernel<<<combGrid, blk, 0, stream>>>(bufB, bufA, dinv, b2, GNN_N, 1);

    // ---- layer 3: gcn(bufB, W3, b3) -> bufB (no relu) ----
    gnn_gemm64_kernel<<<gemmGrid, blk, 0, stream>>>(bufB, W3, bufA);
    hipMemsetAsync(bufB, 0, featBytes, stream);
    gnn_scatter_kernel<<<scatGrid, blk, 0, stream>>>(bufA, bufB, src, dst, dinv, GNN_E);
    gnn_combine_kernel<<<combGrid, blk, 0, stream>>>(bufB, bufA, dinv, b3, GNN_N, 0);

    // ---- global mean pool -> d_out [G,64] ----
    hipMemsetAsync(sums, 0, ((size_t)GNN_G * GNN_H + GNN_G) * sizeof(float), stream);
    gnn_pool_kernel<<<dim3((unsigned)(((long long)GNN_N * 32 + TB - 1) / TB)), blk, 0, stream>>>(
        bufB, batch, sums, cnt, GNN_N);
    gnn_final_kernel<<<dim3((GNN_G * GNN_H + TB - 1) / TB), blk, 0, stream>>>(
        sums, cnt, (float*)d_out, GNN_G);
}